// CVAEEncoder_38268158607905
// MI455X (gfx1250) — compile-verified
//
#include <hip/hip_runtime.h>
#include <math.h>

#define NN 100000
#define NE 600000
#define FI 128
#define HD 128
#define ZD 64

typedef __attribute__((ext_vector_type(2))) float v2f;
typedef __attribute__((ext_vector_type(8))) float v8f;

__global__ __launch_bounds__(256) void zero_kernel(float* __restrict__ p, int n) {
    int i = blockIdx.x * 256 + threadIdx.x;
    if (i < n) p[i] = 0.0f;
}

__global__ __launch_bounds__(256) void deg_kernel(const int* __restrict__ s,
                                                  const int* __restrict__ r,
                                                  float* __restrict__ sdeg,
                                                  float* __restrict__ rdeg) {
    int i = blockIdx.x * 256 + threadIdx.x;
    if (i < NE) {
        atomicAdd(&sdeg[s[i]], 1.0f);
        atomicAdd(&rdeg[r[i]], 1.0f);
    }
}

// One wave per edge: gather a 128-float row from T[sender], atomically add into AGG[receiver].
__global__ __launch_bounds__(256) void scatter_kernel(const int* __restrict__ s,
                                                      const int* __restrict__ r,
                                                      const float* __restrict__ T,
                                                      float* __restrict__ AGG) {
    int e = blockIdx.x * 8 + (threadIdx.x >> 5);
    if (e >= NE) return;
    int lane = threadIdx.x & 31;
    int si = s[e], ri = r[e];
    const float4 v = *reinterpret_cast<const float4*>(T + (size_t)si * HD + lane * 4);
    float* dst = AGG + (size_t)ri * HD + lane * 4;
    atomicAdd(dst + 0, v.x);
    atomicAdd(dst + 1, v.y);
    atomicAdd(dst + 2, v.z);
    atomicAdd(dst + 3, v.w);
}

// One block = 16 node rows. 8 waves, each owns a 16-wide N-tile of the 128-wide output.
// Computes T[row] = softmax(relu(Xs @ W + b)) * rsqrt(max(sdeg,1)), where Xs optionally
// pre-scales each input row by rsqrt(max(rdeg,1)) (folded receiver normalization).
__global__ __launch_bounds__(256) void gc_dense_kernel(
    const float* __restrict__ X, const float* __restrict__ rdeg,
    const float* __restrict__ W, const float* __restrict__ bias,
    const float* __restrict__ sdeg, float* __restrict__ T) {
    __shared__ float lds[16 * 132];  // stride 132 to avoid bank conflicts
    const int tid = threadIdx.x;
    const int lane = tid & 31;
    const int wave = tid >> 5;
    const int rowBase = blockIdx.x * 16;

    // Stage 16x128 A tile into LDS (2048 floats, 8 per thread as two float4s)
#pragma unroll
    for (int halfT = 0; halfT < 2; ++halfT) {
        int g = halfT * 1024 + tid * 4;
        int rrow = g >> 7;
        int c = g & 127;
        float4 v = *reinterpret_cast<const float4*>(X + (size_t)(rowBase + rrow) * FI + c);
        if (rdeg != nullptr) {
            float sc = rsqrtf(fmaxf(rdeg[rowBase + rrow], 1.0f));
            v.x *= sc; v.y *= sc; v.z *= sc; v.w *= sc;
        }
        *reinterpret_cast<float4*>(&lds[rrow * 132 + c]) = v;
    }
    __syncthreads();

    const int half = lane >> 4;   // half-wave select (K pair)
    const int ml = lane & 15;     // M for A, N-within-tile for B/C
    const int ncol = wave * 16 + ml;
    v8f acc = {0, 0, 0, 0, 0, 0, 0, 0};
    for (int k = 0; k < FI; k += 4) {
        const int kk = k + 2 * half;
        // A 16x4 f32 layout: lane m holds A[m][kk], A[m][kk+1]
        v2f a = *reinterpret_cast<const v2f*>(&lds[ml * 132 + kk]);
        // B 4x16 f32 layout: lane n holds W[kk][n], W[kk+1][n]
        v2f b;
        b.x = W[(size_t)kk * HD + ncol];
        b.y = W[(size_t)(kk + 1) * HD + ncol];
        acc = __builtin_amdgcn_wmma_f32_16x16x4_f32(false, a, false, b, (short)0, acc,
                                                    false, false);
    }
    __syncthreads();
    // C/D layout: VGPR g -> rows g (lanes 0-15) and g+8 (lanes 16-31). bias + relu into LDS.
#pragma unroll
    for (int g = 0; g < 8; ++g) {
        int rrow = g + 8 * half;
        lds[rrow * 132 + ncol] = fmaxf(acc[g] + bias[ncol], 0.0f);
    }
    __syncthreads();

    // Row softmax (128 wide) + sender-degree scale; each wave handles 2 rows, 4 elems/lane.
#pragma unroll
    for (int rr = 0; rr < 2; ++rr) {
        const int rrow = wave * 2 + rr;
        float4 v = *reinterpret_cast<const float4*>(&lds[rrow * 132 + lane * 4]);
        float m = fmaxf(fmaxf(v.x, v.y), fmaxf(v.z, v.w));
#pragma unroll
        for (int off = 16; off >= 1; off >>= 1) m = fmaxf(m, __shfl_xor(m, off, 32));
        float e0 = __expf(v.x - m), e1 = __expf(v.y - m);
        float e2 = __expf(v.z - m), e3 = __expf(v.w - m);
        float s = e0 + e1 + e2 + e3;
#pragma unroll
        for (int off = 16; off >= 1; off >>= 1) s += __shfl_xor(s, off, 32);
        const float sc = rsqrtf(fmaxf(sdeg[rowBase + rrow], 1.0f)) / s;
        float4 o;
        o.x = e0 * sc; o.y = e1 * sc; o.z = e2 * sc; o.w = e3 * sc;
        *reinterpret_cast<float4*>(T + (size_t)(rowBase + rrow) * HD + lane * 4) = o;
    }
}

// Head: hcat = [AGG*rsqrt(max(rdeg,1)), nodes] (256 wide), mu = hcat@Wmu+bmu, ls = hcat@Wls+bls.
// Waves 0-3 -> mu N-tiles, waves 4-7 -> logsig2 N-tiles.
__global__ __launch_bounds__(256) void head_kernel(
    const float* __restrict__ AGG, const float* __restrict__ rdeg,
    const float* __restrict__ nodes,
    const float* __restrict__ Wmu, const float* __restrict__ bmu,
    const float* __restrict__ Wls, const float* __restrict__ bls,
    float* __restrict__ out_mu, float* __restrict__ out_ls) {
    __shared__ float lds[16 * 260];  // 16 rows x 256 cols, stride 260
    const int tid = threadIdx.x;
    const int lane = tid & 31;
    const int wave = tid >> 5;
    const int rowBase = blockIdx.x * 16;

#pragma unroll
    for (int q = 0; q < 4; ++q) {
        int g = q * 1024 + tid * 4;
        int rrow = g >> 8;
        int c = g & 255;
        float4 v;
        if (c < 128) {
            v = *reinterpret_cast<const float4*>(AGG + (size_t)(rowBase + rrow) * HD + c);
            float sc = rsqrtf(fmaxf(rdeg[rowBase + rrow], 1.0f));
            v.x *= sc; v.y *= sc; v.z *= sc; v.w *= sc;
        } else {
            v = *reinterpret_cast<const float4*>(nodes + (size_t)(rowBase + rrow) * FI +
                                                 (c - 128));
        }
        *reinterpret_cast<float4*>(&lds[rrow * 260 + c]) = v;
    }
    __syncthreads();

    const int half = lane >> 4;
    const int ml = lane & 15;
    const int mat = wave >> 2;  // 0 = mu, 1 = logsig2
    const int ncol = (wave & 3) * 16 + ml;
    const float* __restrict__ Wsel = mat ? Wls : Wmu;
    const float* __restrict__ bsel = mat ? bls : bmu;
    float* __restrict__ osel = mat ? out_ls : out_mu;

    v8f acc = {0, 0, 0, 0, 0, 0, 0, 0};
    for (int k = 0; k < HD + FI; k += 4) {
        const int kk = k + 2 * half;
        v2f a = *reinterpret_cast<const v2f*>(&lds[ml * 260 + kk]);
        v2f b;
        b.x = Wsel[(size_t)kk * ZD + ncol];
        b.y = Wsel[(size_t)(kk + 1) * ZD + ncol];
        acc = __builtin_amdgcn_wmma_f32_16x16x4_f32(false, a, false, b, (short)0, acc,
                                                    false, false);
    }
    const float bb = bsel[ncol];
#pragma unroll
    for (int g = 0; g < 8; ++g) {
        int rrow = g + 8 * half;
        osel[(size_t)(rowBase + rrow) * ZD + ncol] = acc[g] + bb;
    }
}

extern "C" void kernel_launch(void* const* d_in, const int* in_sizes, int n_in,
                              void* d_out, int out_size, void* d_ws, size_t ws_size,
                              hipStream_t stream) {
    (void)in_sizes; (void)n_in; (void)out_size; (void)ws_size;
    const float* nodes = (const float*)d_in[0];
    const int* senders = (const int*)d_in[1];
    const int* receivers = (const int*)d_in[2];
    const float* W0 = (const float*)d_in[3];
    const float* b0 = (const float*)d_in[4];
    const float* W1 = (const float*)d_in[5];
    const float* b1 = (const float*)d_in[6];
    const float* Wmu = (const float*)d_in[7];
    const float* bmu = (const float*)d_in[8];
    const float* Wls = (const float*)d_in[9];
    const float* bls = (const float*)d_in[10];

    float* out = (float*)d_out;
    float* out_mu = out;
    float* out_ls = out + (size_t)NN * ZD;

    // Workspace: AGG [NN*HD] + sdeg [NN] + rdeg [NN]  (~52 MB)
    float* AGG = (float*)d_ws;
    float* SDEG = AGG + (size_t)NN * HD;
    float* RDEG = SDEG + NN;
    // d_out (NN*ZD*2 == NN*HD floats) doubles as dense-output scratch T;
    // its T-lifetime ends (last read: scatter) before head_kernel overwrites it.
    float* T = out;

    const int nodeBlocks = NN / 16;                 // 6250
    const int edgeBlocks = (NE + 255) / 256;        // 2344
    const int scatterBlocks = (NE + 7) / 8;         // 75000
    const int featN = NN * HD;                      // 12.8M
    const int featBlocks = (featN + 255) / 256;

    // Degrees
    zero_kernel<<<(2 * NN + 255) / 256, 256, 0, stream>>>(SDEG, 2 * NN);
    deg_kernel<<<edgeBlocks, 256, 0, stream>>>(senders, receivers, SDEG, RDEG);

    // Layer 0
    gc_dense_kernel<<<nodeBlocks, 256, 0, stream>>>(nodes, nullptr, W0, b0, SDEG, T);
    zero_kernel<<<featBlocks, 256, 0, stream>>>(AGG, featN);
    scatter_kernel<<<scatterBlocks, 256, 0, stream>>>(senders, receivers, T, AGG);

    // Layer 1 (receiver scaling of layer-0 output folded into A-tile load)
    gc_dense_kernel<<<nodeBlocks, 256, 0, stream>>>(AGG, RDEG, W1, b1, SDEG, T);
    zero_kernel<<<featBlocks, 256, 0, stream>>>(AGG, featN);
    scatter_kernel<<<scatterBlocks, 256, 0, stream>>>(senders, receivers, T, AGG);

    // Head (receiver scaling folded; concat built in LDS)
    head_kernel<<<nodeBlocks, 256, 0, stream>>>(AGG, RDEG, nodes, Wmu, bmu, Wls, bls,
                                                out_mu, out_ls);
}